// CenterLoss_41077067219594
// MI455X (gfx1250) — compile-verified
//
#include <hip/hip_runtime.h>
#include <hip/hip_bf16.h>

// CenterLoss on MI455X (gfx1250, wave32).
// Only the labeled column of the 8192x50000 distmat survives the mask, so we
// gather the labeled center per row and compute d_i = ||x||^2 + ||c||^2 - 2 x.c
// (same compute order as the reference). The 49,999 masked zeros per row clamp
// to 1e-12 -> closed-form constant (C_OUT-1)*1e-12 added at the end.
// Dot products run on the fp32 matrix pipe: V_WMMA_F32_16X16X4_F32, one wave
// per 16-row tile, diagonal of the Gram tile = the 16 needed dots.

typedef float v8f __attribute__((ext_vector_type(8)));
typedef float v2f __attribute__((ext_vector_type(2)));

#define BS    8192
#define C_OUT 50000
#define DIM   64

__global__ void __launch_bounds__(256)
center_loss_wmma(const float* __restrict__ x,
                 const int*   __restrict__ labels,
                 const float* __restrict__ centers,
                 float*       __restrict__ partials) {
  const int lane  = threadIdx.x & 31;
  const int wave  = threadIdx.x >> 5;
  const int tile  = blockIdx.x * (blockDim.x >> 5) + wave;  // 512 tiles of 16 rows
  const int r     = lane & 15;
  const int row   = tile * 16 + r;
  const int koff  = (lane < 16) ? 0 : 2;   // 32-bit A 16x4 layout: lanes 0-15 hold K=0,1; 16-31 hold K=2,3

  const int lab = labels[row];
  const float* __restrict__ xrow = x + (size_t)row * DIM;
  const float* __restrict__ crow = centers + (size_t)lab * DIM;

  v8f  acc = {};          // C/D accumulator (16x16 f32, 8 VGPRs)
  float sqx = 0.0f;       // partial sum of x^2 for this lane's half-row
  float sqc = 0.0f;       // partial sum of c^2 for this lane's half-row

#pragma unroll
  for (int t = 0; t < DIM / 4; ++t) {       // 16 WMMA steps, K=4 each
    const int k = t * 4 + koff;
    v2f a = *(const v2f*)(xrow + k);        // A: x rows   (16x4 fp32)
    v2f b = *(const v2f*)(crow + k);        // B: gathered centers as 4x16 (same lane mapping)
    sqx += a.x * a.x + a.y * a.y;
    sqc += b.x * b.x + b.y * b.y;
    acc = __builtin_amdgcn_wmma_f32_16x16x4_f32(
        /*neg_a=*/false, a, /*neg_b=*/false, b,
        /*c_mod=*/(short)0, acc, /*reuse_a=*/false, /*reuse_b=*/false);
  }

  // Row r's 64 elements are split between lanes r and r+16 -> one xor-16 combine.
  sqx += __shfl_xor(sqx, 16, 32);
  sqc += __shfl_xor(sqc, 16, 32);

  // Diagonal of D: lane l<8 owns row l in acc[l]; lane l>=24 owns row l-16 in acc[l-24].
  const bool owns = (lane < 8) || (lane >= 24);
  const int  idx  = (lane < 8) ? lane : (lane - 24);
  float diag = acc[0];
#pragma unroll
  for (int v = 1; v < 8; ++v) diag = (idx == v) ? acc[v] : diag;

  float dist = sqx + sqc - 2.0f * diag;                 // reference compute order
  dist = fminf(fmaxf(dist, 1e-12f), 1e12f);             // clamp AFTER masking
  float contrib = owns ? dist : 0.0f;

  // Fixed-shape wave32 reduction (deterministic).
#pragma unroll
  for (int s = 16; s >= 1; s >>= 1) contrib += __shfl_xor(contrib, s, 32);

  if (lane == 0) partials[tile] = contrib;
}

__global__ void __launch_bounds__(512)
reduce_partials(const float* __restrict__ partials, float* __restrict__ out) {
  __shared__ float smem[512];
  const int t = threadIdx.x;
  smem[t] = partials[t];
  __syncthreads();
#pragma unroll
  for (int s = 256; s >= 1; s >>= 1) {
    if (t < s) smem[t] += smem[t + s];
    __syncthreads();
  }
  if (t == 0) {
    // masked-out entries: BS*(C_OUT-1) zeros clamp to 1e-12, then /BS.
    out[0] = smem[0] / (float)BS + (float)(C_OUT - 1) * 1e-12f;
  }
}

extern "C" void kernel_launch(void* const* d_in, const int* in_sizes, int n_in,
                              void* d_out, int out_size, void* d_ws, size_t ws_size,
                              hipStream_t stream) {
  const float* x       = (const float*)d_in[0];
  const int*   labels  = (const int*)  d_in[1];
  const float* centers = (const float*)d_in[2];
  float*       out     = (float*)d_out;
  float*       parts   = (float*)d_ws;   // 512 floats = 2 KB

  // 512 tiles, 8 waves (256 threads) per block -> 64 blocks.
  center_loss_wmma<<<64, 256, 0, stream>>>(x, labels, centers, parts);
  reduce_partials<<<1, 512, 0, stream>>>(parts, out);
}